// LineNNUEv2_83640193122477
// MI455X (gfx1250) — compile-verified
//
#include <hip/hip_runtime.h>
#include <math.h>

// ---------------------------------------------------------------------------
// LineNNUE v2 forward, MI455X (gfx1250, wave32).
//   Kernel A: per-board fused gather+tanh+mean+PReLU -> LDS -> dwconv+ReLU
//             -> fused policy head + fused global mean pool.
//   Kernel B: value MLP (32->32->32->3) using V_WMMA_F32_16X16X4_F32.
// ---------------------------------------------------------------------------

#define BQ   4096
#define LL   4
#define HH   15
#define WW   15
#define NPOS (HH * WW)        // 225
#define CC   32
#define MAPMAX 32.0f

typedef __attribute__((ext_vector_type(2))) float v2f;
typedef __attribute__((ext_vector_type(8))) float v8f;

// ===========================================================================
// Kernel A: one block per board. 256 threads = 8 waves; lane = channel.
// ===========================================================================
__global__ __launch_bounds__(256) void fused_front_kernel(
    const int*   __restrict__ enc,          // (B, 4, 15, 15)
    const float* __restrict__ emb,          // (NUM_CODES, 32)
    const float* __restrict__ prelu_map,    // (32,)
    const float* __restrict__ dwconv_w,     // (32, 1, 3, 3)
    const float* __restrict__ dwconv_b,     // (32,)
    const float* __restrict__ policy_w,     // (1, 16, 1, 1)
    const float* __restrict__ prelu_policy, // (1,)
    float*       __restrict__ policy_out,   // (B, 225)
    float*       __restrict__ vpool)        // (B, 32) workspace
{
  __shared__ float xs[NPOS * CC];   // 28.8 KB: post-PReLU features, NHWC
  __shared__ float red[8 * CC];     // cross-wave value-pool reduction

  const int b    = blockIdx.x;
  const int tid  = threadIdx.x;
  const int c    = tid & 31;        // channel == lane
  const int wave = tid >> 5;

  // ---- stage 1: gather + scaled-tanh + mean over L + per-channel PReLU ----
  const float amap  = prelu_map[c];
  const int*  ebase = enc + (size_t)b * LL * NPOS;
  for (int i = tid; i < NPOS * CC; i += 256) {
    const int pos = i >> 5;                       // i & 31 == c
    float acc = 0.f;
#pragma unroll
    for (int l = 0; l < LL; ++l) {
      const int idx = ebase[l * NPOS + pos];      // wave-uniform (scalar load)
      const float e = emb[(size_t)idx * CC + c];  // 128B coalesced per wave
      acc += MAPMAX * tanhf(e * (1.0f / MAPMAX));
    }
    float f = acc * 0.25f;                        // mean over L=4
    xs[i] = (f >= 0.f) ? f : amap * f;            // PReLU(C)
  }
  __syncthreads();

  // ---- stage 2: depthwise 3x3 + bias + ReLU, fused policy & value heads ---
  float wreg[9];
#pragma unroll
  for (int j = 0; j < 9; ++j) wreg[j] = dwconv_w[c * 9 + j];
  const float bias = dwconv_b[c];
  const float pw   = (c < 16) ? policy_w[c] : 0.f;
  const float pa   = prelu_policy[0];

  float vacc = 0.f;
  for (int it = 0; it < 29; ++it) {
    const int p = it * 8 + wave;                  // wave-uniform position
    float contrib = 0.f;
    if (p < NPOS) {
      const int h = p / WW, w = p % WW;
      float a = bias;
#pragma unroll
      for (int kh = 0; kh < 3; ++kh) {
        const int hh = h + kh - 1;
        if ((unsigned)hh < (unsigned)HH) {
#pragma unroll
          for (int kw = 0; kw < 3; ++kw) {
            const int ww2 = w + kw - 1;
            if ((unsigned)ww2 < (unsigned)WW)
              a += xs[(hh * WW + ww2) * CC + c] * wreg[kh * 3 + kw];
          }
        }
      }
      const float y = fmaxf(a, 0.f);              // ReLU
      vacc += y;                                  // value pool accumulate
      contrib = y * pw;                           // policy 1x1 (c<16 only)
    }
    // wave32 all-lane reduce of policy dot product
#pragma unroll
    for (int off = 16; off > 0; off >>= 1)
      contrib += __shfl_xor(contrib, off, 32);
    if (p < NPOS && c == 0) {
      const float pol = (contrib >= 0.f) ? contrib : pa * contrib;
      policy_out[(size_t)b * NPOS + p] = pol;
    }
  }

  // ---- cross-wave value-pool reduction ----
  red[wave * CC + c] = vacc;
  __syncthreads();
  if (tid < CC) {
    float s = 0.f;
#pragma unroll
    for (int w8 = 0; w8 < 8; ++w8) s += red[w8 * CC + tid];
    vpool[(size_t)b * CC + tid] = s * (1.0f / (float)NPOS);
  }
}

// ===========================================================================
// Kernel B: value MLP with V_WMMA_F32_16X16X4_F32.
// Each wave owns 16 batch rows; 2 N-tiles x 8 K-steps per 32x32 layer.
// ===========================================================================
__device__ __forceinline__ v8f wmma4(v2f a, v2f b, v8f c) {
  // 8 args: (neg_a, A, neg_b, B, c_mod, C, reuse_a, reuse_b)
  return __builtin_amdgcn_wmma_f32_16x16x4_f32(
      false, a, false, b, (short)0, c, false, false);
}

#define HSTRIDE 34  // LDS row stride: lane halves hit disjoint bank groups

__global__ __launch_bounds__(256) void value_mlp_kernel(
    const float* __restrict__ vpool,  // (B, 32)
    const float* __restrict__ w1, const float* __restrict__ b1,
    const float* __restrict__ w2, const float* __restrict__ b2,
    const float* __restrict__ w3, const float* __restrict__ b3,
    float*       __restrict__ out)    // (B, 3)
{
  __shared__ float hb[8][16 * HSTRIDE];  // per-wave 16x32 transpose staging

  const int tid   = threadIdx.x;
  const int wave  = tid >> 5;
  const int lane  = tid & 31;
  const int lh    = lane & 15;           // M (A) / N (B,C) sub-index
  const int kh    = (lane >> 4) * 2;     // K offset: 0 (lanes 0-15) / 2 (16-31)
  const int row0  = blockIdx.x * 128 + wave * 16;
  float* hbuf = hb[wave];

  // ---------------- layer 1: h1 = relu(v @ w1^T + b1) ----------------
  v8f c0 = {}, c1 = {};
  {
    const float* arow = vpool + (size_t)(row0 + lh) * 32;
#pragma unroll
    for (int k0 = 0; k0 < 32; k0 += 4) {
      v2f a;  a.x  = arow[k0 + kh];              a.y  = arow[k0 + kh + 1];
      v2f bA; bA.x = w1[lh * 32 + k0 + kh];      bA.y = w1[lh * 32 + k0 + kh + 1];
      v2f bB; bB.x = w1[(lh + 16) * 32 + k0 + kh];
              bB.y = w1[(lh + 16) * 32 + k0 + kh + 1];
      c0 = wmma4(a, bA, c0);
      c1 = wmma4(a, bB, c1);
    }
  }
  // bias + relu, transpose through LDS (C layout: lane=N, vgpr v -> M)
  {
    const float bb0 = b1[lh], bb1 = b1[lh + 16];
#pragma unroll
    for (int v = 0; v < 8; ++v) {
      const int m = v + (lane >> 4) * 8;
      hbuf[m * HSTRIDE + lh]      = fmaxf(c0[v] + bb0, 0.f);
      hbuf[m * HSTRIDE + lh + 16] = fmaxf(c1[v] + bb1, 0.f);
    }
  }
  __syncthreads();

  // ---------------- layer 2: h2 = relu(h1 @ w2^T + b2) ----------------
  v8f d0 = {}, d1 = {};
  {
    const float* arow = hbuf + lh * HSTRIDE;
#pragma unroll
    for (int k0 = 0; k0 < 32; k0 += 4) {
      v2f a;  a.x  = arow[k0 + kh];              a.y  = arow[k0 + kh + 1];
      v2f bA; bA.x = w2[lh * 32 + k0 + kh];      bA.y = w2[lh * 32 + k0 + kh + 1];
      v2f bB; bB.x = w2[(lh + 16) * 32 + k0 + kh];
              bB.y = w2[(lh + 16) * 32 + k0 + kh + 1];
      d0 = wmma4(a, bA, d0);
      d1 = wmma4(a, bB, d1);
    }
  }
  __syncthreads();   // layer-1 reads done before overwriting hbuf
  {
    const float bb0 = b2[lh], bb1 = b2[lh + 16];
#pragma unroll
    for (int v = 0; v < 8; ++v) {
      const int m = v + (lane >> 4) * 8;
      hbuf[m * HSTRIDE + lh]      = fmaxf(d0[v] + bb0, 0.f);
      hbuf[m * HSTRIDE + lh + 16] = fmaxf(d1[v] + bb1, 0.f);
    }
  }
  __syncthreads();

  // ---------------- layer 3: out = h2 @ w3^T + b3  (32 -> 3) ----------------
  if (lane < 16) {
    const float* hrow = hbuf + lh * HSTRIDE;
    float a0 = b3[0], a1 = b3[1], a2 = b3[2];
#pragma unroll
    for (int k = 0; k < 32; ++k) {
      const float h = hrow[k];
      a0 += h * w3[k];
      a1 += h * w3[32 + k];
      a2 += h * w3[64 + k];
    }
    const size_t r = (size_t)(row0 + lh) * 3;
    out[r + 0] = a0; out[r + 1] = a1; out[r + 2] = a2;
  }
}

// ===========================================================================
extern "C" void kernel_launch(void* const* d_in, const int* in_sizes, int n_in,
                              void* d_out, int out_size, void* d_ws, size_t ws_size,
                              hipStream_t stream) {
  (void)in_sizes; (void)n_in; (void)out_size; (void)ws_size;
  const int*   enc          = (const int*)  d_in[0];
  const float* emb          = (const float*)d_in[1];
  const float* prelu_map    = (const float*)d_in[2];
  const float* dwconv_w     = (const float*)d_in[3];
  const float* dwconv_b     = (const float*)d_in[4];
  const float* policy_w     = (const float*)d_in[5];
  const float* prelu_policy = (const float*)d_in[6];
  const float* w1 = (const float*)d_in[7];
  const float* b1 = (const float*)d_in[8];
  const float* w2 = (const float*)d_in[9];
  const float* b2 = (const float*)d_in[10];
  const float* w3 = (const float*)d_in[11];
  const float* b3 = (const float*)d_in[12];

  float* out        = (float*)d_out;
  float* value_out  = out;                       // (4096, 3)
  float* policy_out = out + (size_t)BQ * 3;      // (4096, 225)
  float* vpool      = (float*)d_ws;              // (4096, 32) = 512 KB scratch

  fused_front_kernel<<<BQ, 256, 0, stream>>>(
      enc, emb, prelu_map, dwconv_w, dwconv_b, policy_w, prelu_policy,
      policy_out, vpool);

  value_mlp_kernel<<<BQ / 128, 256, 0, stream>>>(
      vpool, w1, b1, w2, b2, w3, b3, value_out);
}